// TsodycsMarkramSynapse_49624052138065
// MI455X (gfx1250) — compile-verified
//
#include <hip/hip_runtime.h>

// ---------------------------------------------------------------------------
// Tsodyks-Markram synapse bank, MI455X (gfx1250, wave32).
//
// - Scan carry (R,U,A) is elementwise in (b,n): integrate time in registers.
// - Reductions over N are output-only: per wave, stage gsyn[16t x 64n] in LDS
//   and compute both sums (g_tot, sum g*Erev) for 16 timesteps with
//   V_WMMA_F32_16X16X4_F32; waves write disjoint per-wave partial slices
//   (plain ds stores, no atomics -> bit-deterministic), summed at the end.
// - x streamed via async global->LDS DMA (ASYNCcnt), double-buffered two
//   tiles ahead, so the sequential state chain never waits on HBM.
// Bandwidth-bound: 128 MiB of x @ 23.3 TB/s ~= 5.8 us floor.
// ---------------------------------------------------------------------------

namespace {
constexpr int kB       = 32;
constexpr int kT       = 128;
constexpr int kN       = 8192;
constexpr int kChunks  = 16;               // N-chunks (blocks per batch)
constexpr int kCN      = kN / kChunks;     // 512 synapses per block
constexpr int kThreads = 256;              // 8 waves
constexpr int kWaves   = kThreads / 32;
constexpr int kNPerWv  = kCN / kWaves;     // 64 synapses per wave (2/lane)
constexpr int kTT      = 16;               // time tile (== WMMA M)
constexpr int kTiles   = kT / kTT;
constexpr int kGRow    = 68;               // gsyn LDS row stride (bank-conflict free)
constexpr float kDT    = 0.1f;
}

typedef __attribute__((ext_vector_type(2))) float v2f;
typedef __attribute__((ext_vector_type(8))) float v8f;
typedef __attribute__((ext_vector_type(4))) int   v4i;

#if defined(__AMDGCN__) && __has_builtin(__builtin_amdgcn_wmma_f32_16x16x4_f32)
#define TM_USE_WMMA 1
#else
#define TM_USE_WMMA 0
#endif

#if defined(__AMDGCN__) && \
    __has_builtin(__builtin_amdgcn_global_load_async_to_lds_b128) && \
    __has_builtin(__builtin_amdgcn_s_wait_asynccnt)
#define TM_ASYNC 1
#else
#define TM_ASYNC 0
#endif

#if TM_ASYNC
// Builtin expects (AS1 v4i* global_src, AS3 v4i* lds_dst, imm offset, imm cpol).
#define TM_GPTR(p) ((__attribute__((address_space(1))) v4i*)(uintptr_t)(p))
#define TM_LPTR(p) ((__attribute__((address_space(3))) v4i*)(uint32_t)(uintptr_t)(p))
#endif

__global__ __launch_bounds__(kThreads)
void tm_state_kernel(const float* __restrict__ x,
                     const float* __restrict__ pconn,
                     const float* __restrict__ erev,
                     const float* __restrict__ gsyn_max,
                     const float* __restrict__ tau_f,
                     const float* __restrict__ tau_d,
                     const float* __restrict__ tau_r,
                     const float* __restrict__ uinc,
                     float* __restrict__ ws)
{
    __shared__ __align__(16) float s_gs[kWaves * kTT * kGRow]; // per-wave gsyn tiles
    __shared__ float s_erev[kCN];
    // Per-wave partial sums [wave][t][{g_tot, sum g*E}]; every entry written
    // exactly once (tiles partition t) -> no init, no atomics, deterministic.
    __shared__ float s_accw[kWaves][kT * 2];
#if TM_ASYNC
    __shared__ __align__(16) float s_xb[2][kTT][kCN]; // async x double buffer
#endif

    const int tid  = threadIdx.x;
    const int wave = tid >> 5;
    const int lane = tid & 31;
    const int b     = blockIdx.x % kB;
    const int chunk = blockIdx.x / kB;
    const int nbase = chunk * kCN;                 // base index in N for block
    const int nloc  = wave * kNPerWv + lane * 2;   // chunk-local index (2/lane)
    const int n0    = nbase + nloc;

    // ---- per-synapse parameters (2 per lane), precompute decay factors ----
    const float pc0 = pconn[n0],     pc1 = pconn[n0 + 1];
    const float E0  = erev[n0],      E1  = erev[n0 + 1];
    const float gm0 = fmaxf(gsyn_max[n0],     0.0f);
    const float gm1 = fmaxf(gsyn_max[n0 + 1], 0.0f);
    const float td0 = tau_d[n0], td1 = tau_d[n0 + 1];
    const float tr0 = tau_r[n0], tr1 = tau_r[n0 + 1];
    const float tf0 = tau_f[n0], tf1 = tau_f[n0 + 1];
    const float ui0 = uinc[n0],  ui1 = uinc[n0 + 1];

    const float ed0  = expf(-kDT / td0), ed1  = expf(-kDT / td1);
    const float ef0  = expf(-kDT / tf0), ef1  = expf(-kDT / tf1);
    const float exr0 = expf(-kDT / tr0), exr1 = expf(-kDT / tr1);
    const float t1r0 = (td0 != tr0) ? td0 / (td0 - tr0) : 1e-13f;
    const float t1r1 = (td1 != tr1) ? td1 / (td1 - tr1) : 1e-13f;

    s_erev[nloc]     = E0;
    s_erev[nloc + 1] = E1;

    const float* xblk  = x + (size_t)b * kT * kN + nbase;
    float*       gsrow = s_gs + wave * (kTT * kGRow);

#if TM_ASYNC
    // Issue one 16-row x tile (16 x 512 floats) as 2048 b128 async DMAs.
    auto issue_tile = [&](int tile, int buf) {
        const int t0 = tile * kTT;
        #pragma unroll
        for (int j = 0; j < (kTT * kCN / 4) / kThreads; ++j) { // 8 per thread
            const int s    = tid + j * kThreads;  // 0..2047
            const int row  = s >> 7;              // 0..15
            const int col  = (s & 127) << 2;      // float column, 16B aligned
            const float* g = xblk + (size_t)(t0 + row) * kN + col;
            float*       l = &s_xb[buf][row][col];
            __builtin_amdgcn_global_load_async_to_lds_b128(TM_GPTR(g), TM_LPTR(l),
                                                           0, 0);
        }
    };
    issue_tile(0, 0);
    issue_tile(1, 1);
#endif

    // ---- state registers ----
    float R0 = 1.0f, U0 = 0.0f, A0 = 0.0f;
    float R1 = 1.0f, U1 = 0.0f, A1 = 0.0f;

#if TM_USE_WMMA
    const int trow = lane & 15;        // A: M (time-in-tile) for this lane
    const int npos = lane & 15;        // B/D: column index for this lane
    const int half = lane >> 4;
    const int koff = half * 2;         // A/B: K offset {0,2} by lane half

    // Hoist all 16 B operands (col0 = ones, col1 = Erev, rest 0) into regs.
    v2f bvs[kNPerWv / 4];
    #pragma unroll
    for (int kb4 = 0; kb4 < kNPerWv / 4; ++kb4) {
        const float2 ev = *reinterpret_cast<const float2*>(
            &s_erev[wave * kNPerWv + kb4 * 4 + koff]);
        v2f bv;
        bv.x = (npos == 0) ? 1.0f : ((npos == 1) ? ev.x : 0.0f);
        bv.y = (npos == 0) ? 1.0f : ((npos == 1) ? ev.y : 0.0f);
        bvs[kb4] = bv;
    }
#endif

    for (int tile = 0; tile < kTiles; ++tile) {
        const int t0 = tile * kTT;
#if TM_ASYNC
        const int buf = tile & 1;
        if (tile < kTiles - 1) {
            __builtin_amdgcn_s_wait_asynccnt(8); // this tile's 8 DMAs landed
        } else {
            __builtin_amdgcn_s_wait_asynccnt(0);
        }
        __syncthreads(); // all waves' DMA writes visible
#endif

        // -------- elementwise integration for 16 steps; stash gsyn --------
        #pragma unroll
        for (int tt = 0; tt < kTT; ++tt) {
#if TM_ASYNC
            const float2 xv =
                *reinterpret_cast<const float2*>(&s_xb[buf][tt][nloc]);
#else
            const int t = t0 + tt;
            const float* xp = xblk + (size_t)t * kN + nloc;
            if (lane == 0 && t + kTT < kT) { // prefetch next tile's row span
                __builtin_prefetch(xp + kTT * kN, 0, 0);
                __builtin_prefetch(xp + kTT * kN + 32, 0, 0);
            }
            const float2 xv = *reinterpret_cast<const float2*>(xp);
#endif
            // synapse 0
            float frn = xv.x * pc0;
            float a_  = A0 * ed0;
            float r_  = 1.0f + (R0 - 1.0f + t1r0 * A0) * exr0 - t1r0 * A0;
            float u_  = U0 * ef0;
            U0 = u_ + ui0 * (1.0f - u_) * frn;
            float urf = U0 * r_ * frn;
            A0 = a_ + urf;
            R0 = r_ - urf;
            const float g0 = gm0 * A0;

            // synapse 1
            frn = xv.y * pc1;
            a_  = A1 * ed1;
            r_  = 1.0f + (R1 - 1.0f + t1r1 * A1) * exr1 - t1r1 * A1;
            u_  = U1 * ef1;
            U1 = u_ + ui1 * (1.0f - u_) * frn;
            urf = U1 * r_ * frn;
            A1 = a_ + urf;
            R1 = r_ - urf;
            const float g1 = gm1 * A1;

#if TM_USE_WMMA
            *reinterpret_cast<float2*>(&gsrow[tt * kGRow + 2 * lane]) =
                make_float2(g0, g1);
#else
            // fallback: shuffle-tree reduction per step, wave-private store
            float gt = g0 + g1;
            float nm = g0 * E0 + g1 * E1;
            #pragma unroll
            for (int off = 16; off; off >>= 1) {
                gt += __shfl_xor(gt, off, 32);
                nm += __shfl_xor(nm, off, 32);
            }
            if (lane == 0) {
                s_accw[wave][(t0 + tt) * 2 + 0] = gt;
                s_accw[wave][(t0 + tt) * 2 + 1] = nm;
            }
#endif
        }

#if TM_USE_WMMA
        // -------- wave-local WMMA reduction over 64 synapses ---------------
        // D(16x16) += A(16x4: gsyn[t, n..n+3]) x B(4x16: col0=1, col1=Erev)
        // A layout: lanes0-15 M=lane K={0,1}; lanes16-31 M=lane-16 K={2,3}
        // B layout: lanes0-15 N=lane K={0,1}; lanes16-31 N=lane-16 K={2,3}
        v8f d = {0.f, 0.f, 0.f, 0.f, 0.f, 0.f, 0.f, 0.f};
        #pragma unroll
        for (int kb = 0; kb < kNPerWv; kb += 4) {
            const v2f a =
                *reinterpret_cast<const v2f*>(&gsrow[trow * kGRow + kb + koff]);
            d = __builtin_amdgcn_wmma_f32_16x16x4_f32(
                    false, a, false, bvs[kb >> 2], (short)0, d, false, false);
        }
        // D: VGPR v, lanes0-15 -> (M=v, N=lane); lanes16-31 -> (M=v+8, N=lane-16)
        // Each (t, col) is produced exactly once per wave -> plain stores.
        if (npos < 2) {
            #pragma unroll
            for (int v = 0; v < 8; ++v) {
                s_accw[wave][(t0 + v + 8 * half) * 2 + npos] = d[v];
            }
        }
#endif

#if TM_ASYNC
        __syncthreads(); // everyone done reading s_xb[buf]
        if (tile + 2 < kTiles) issue_tile(tile + 2, buf);
#endif
    }

    __syncthreads();
    // Deterministic cross-wave combine: fixed-order sum of the 8 wave slices.
    float acc = 0.0f;
    #pragma unroll
    for (int w = 0; w < kWaves; ++w) acc += s_accw[w][tid];
    ws[((size_t)chunk * kB + b) * (kT * 2) + tid] = acc;
}

__global__ __launch_bounds__(256)
void tm_finalize_kernel(const float* __restrict__ ws, float* __restrict__ out)
{
    const int idx = blockIdx.x * blockDim.x + threadIdx.x;
    if (idx >= kB * kT) return;
    const int b = idx / kT;
    const int t = idx % kT;
    float gt = 0.0f, nm = 0.0f;
    #pragma unroll
    for (int c = 0; c < kChunks; ++c) {
        const float* p = ws + ((size_t)c * kB + b) * (kT * 2) + t * 2;
        gt += p[0];
        nm += p[1];
    }
    float E = nm / gt;
    E = (E + 75.0f) * (1.0f / 75.0f);        // (E - EREV_MIN)/(EREV_MAX-EREV_MIN)
    const float tau = logf(1.0f / gt + 1.0f); // log(CM/g_tot + 1), CM = 1
    *reinterpret_cast<float2*>(out + (size_t)idx * 2) = make_float2(E, tau);
}

extern "C" void kernel_launch(void* const* d_in, const int* in_sizes, int n_in,
                              void* d_out, int out_size, void* d_ws, size_t ws_size,
                              hipStream_t stream)
{
    const float* x        = (const float*)d_in[0];
    const float* pconn    = (const float*)d_in[1];
    const float* erev     = (const float*)d_in[2];
    const float* gsyn_max = (const float*)d_in[3];
    const float* tau_f    = (const float*)d_in[4];
    const float* tau_d    = (const float*)d_in[5];
    const float* tau_r    = (const float*)d_in[6];
    const float* uinc     = (const float*)d_in[7];
    float* ws  = (float*)d_ws;   // needs kChunks*kB*kT*2*4 = 512 KiB
    float* out = (float*)d_out;  // [B, T, 2]

    tm_state_kernel<<<kB * kChunks, kThreads, 0, stream>>>(
        x, pconn, erev, gsyn_max, tau_f, tau_d, tau_r, uinc, ws);

    const int total = kB * kT;
    tm_finalize_kernel<<<(total + 255) / 256, 256, 0, stream>>>(ws, out);
}